// gather2dc_67594195304879
// MI455X (gfx1250) — compile-verified
//
#include <hip/hip_runtime.h>
#include <hip/hip_bf16.h>
#include <stdint.h>

// Problem constants (from the reference)
#define KH 16
#define KW 16
#define SH 14
#define SW 14
#define BB 4
#define CC 128
#define HH 256
#define WW 256
#define NIDX 2048

#define CH_PER_BLK 16                      // channels per block
#define THREADS 256                        // 8 waves (wave32)
#define LDS_FLOATS (CH_PER_BLK * KH * KW)  // 4096 floats = 16 KB

// One block = one (patch n, channel-group cg). 8 waves; each wave moves 32
// rows (64B each) global->LDS with async B64 gathers, waits asynccnt==0,
// then streams its own 2KB LDS region to the contiguous output with async
// B128 non-temporal stores. No cross-wave data sharing -> no barriers.
//
// Cache strategy: input (128 MB) fits in the 192 MB L2, so gathered loads use
// default RT policy to stay resident; the 256 MB write-once output stream is
// marked TH_STORE_NT so it does not evict the input from L2.
__global__ __launch_bounds__(THREADS) void gather2dc_async_kernel(
    const float* __restrict__ in, const int* __restrict__ idx,
    float* __restrict__ out) {
  __shared__ float lds[LDS_FLOATS];

  const int blk = blockIdx.x;
  const int n   = blk >> 3;   // patch index (32768 out floats per patch)
  const int cg  = blk & 7;    // channel group of 16

  // Uniform per block -> compiler emits scalar loads
  const int bi = idx[3 * n + 0];
  const int h0 = idx[3 * n + 1] * SH;
  const int w0 = idx[3 * n + 2] * SW;

  const int t    = threadIdx.x;
  const int lane = t & 31;
  const int wv   = t >> 5;
  const int sub  = lane >> 3;        // which of 4 rows this instruction's lane-octet covers
  const int boff = (lane & 7) * 8;   // byte offset within a 64B row

  const float* src_base =
      in + (size_t)bi * ((size_t)CC * HH * WW);  // patch's batch image
  const uint32_t lds_base = (uint32_t)(uintptr_t)&lds[0];

  // ---- Phase 1: async gather 32 rows per wave into LDS -------------------
  // Row r (0..255) = (c_local = r>>4, kh = r&15); LDS row r at bytes [64r,64r+64).
  // Each instruction: lanes 0-7 cover row (4i+0) contiguously, lanes 8-15 row
  // (4i+1), etc -> coalesced 64B segments per lane-octet.
#pragma unroll
  for (int i = 0; i < 8; ++i) {
    const int r  = wv * 32 + i * 4 + sub;          // row within block chunk
    const int c  = cg * CH_PER_BLK + (r >> 4);     // channel
    const int h  = h0 + (r & 15);                  // source row in image
    const uint64_t gaddr =
        (uint64_t)(uintptr_t)(src_base + ((size_t)c * HH + h) * WW + w0) +
        (uint64_t)boff;                            // 8B aligned (w0*4 = wi*56)
    const uint32_t laddr = lds_base + (uint32_t)(r * 64 + boff);
    asm volatile("global_load_async_to_lds_b64 %0, %1, off"
                 :: "v"(laddr), "v"(gaddr)
                 : "memory");
  }

  // Own-wave async ops must complete before async stores read the same LDS.
  asm volatile("s_wait_asynccnt 0x0" ::: "memory");

  // ---- Phase 2: async stream LDS -> contiguous output (non-temporal) -----
  // Block's output chunk: floats [n*32768 + cg*4096, +4096); LDS float j maps
  // 1:1 to chunk float j. Wave wv owns bytes [2048*wv, 2048*(wv+1)).
  float* dst_base = out + ((size_t)n * (CC * KH * KW) +
                           (size_t)cg * (CH_PER_BLK * KH * KW));
#pragma unroll
  for (int i = 0; i < 4; ++i) {
    const uint32_t off   = (uint32_t)(wv * 2048 + i * 512 + lane * 16);  // bytes
    const uint64_t gaddr = (uint64_t)(uintptr_t)dst_base + off;          // 16B aligned
    const uint32_t laddr = lds_base + off;
    asm volatile("global_store_async_from_lds_b128 %0, %1, off th:TH_STORE_NT"
                 :: "v"(gaddr), "v"(laddr)
                 : "memory");
  }

  // S_ENDPGM performs an implicit wait-idle; explicit wait kept for clarity.
  asm volatile("s_wait_asynccnt 0x0" ::: "memory");
}

extern "C" void kernel_launch(void* const* d_in, const int* in_sizes, int n_in,
                              void* d_out, int out_size, void* d_ws,
                              size_t ws_size, hipStream_t stream) {
  (void)in_sizes; (void)n_in; (void)out_size; (void)d_ws; (void)ws_size;
  const float* in  = (const float*)d_in[0];   // (4,128,256,256) fp32
  const int*   idx = (const int*)d_in[1];     // (2048,3) int32
  float*       out = (float*)d_out;           // (2048,128,16,16) fp32

  const int grid = NIDX * (CC / CH_PER_BLK);  // 2048 * 8 = 16384 blocks
  gather2dc_async_kernel<<<grid, THREADS, 0, stream>>>(in, idx, out);
}